// Equalize_20023137533971
// MI455X (gfx1250) — compile-verified
//
#include <hip/hip_runtime.h>
#include <hip/hip_bf16.h>

// ---------------------------------------------------------------------------
// Differentiable histogram equalization for MI455X (gfx1250, wave32).
// Pass 1: truncated soft histogram (LDS ds_add_f32 atomics + v_exp_f32).
// Pass 2: 256-point cumsum as triangular matmul via v_wmma_f32_16x16x4_f32.
// Pass 3: per-pixel equalization; cdfn staged into LDS with
//         global_load_async_to_lds_b32 (ASYNCcnt path) + s_wait_asynccnt.
// ---------------------------------------------------------------------------

typedef float v2f __attribute__((ext_vector_type(2)));
typedef float v8f __attribute__((ext_vector_type(8)));

#define N_BINS   256
#define RADIUS   38                      // exp2(-7213.5*d^2) == 0 in fp32 beyond |k|=37 bins
#define GAUSS_K  7213.4752044448165f     // 5000 * log2(e)
#define INV255   (1.0f / 255.0f)
#define EPS      1e-10f

// ---------------------------------------------------------------------------
// Kernel 1: per-batch soft histogram. grid = (blocksX, 2 batches), 256 thr.
// ---------------------------------------------------------------------------
__global__ __launch_bounds__(256) void hist_kernel(const float* __restrict__ x,
                                                   float* __restrict__ ghist,
                                                   int npix) {
    const int batch = blockIdx.y;
    __shared__ float hs[N_BINS];
    hs[threadIdx.x] = 0.0f;
    __syncthreads();

    const float* xb = x + (size_t)batch * (size_t)npix;
    const int idx    = blockIdx.x * blockDim.x + threadIdx.x;
    const int stride = gridDim.x * blockDim.x;

    for (int i = idx; i < npix; i += stride) {
        if (i + stride < npix) __builtin_prefetch(&xb[i + stride], 0, 0);  // global_prefetch_b8
        const float xv = xb[i];
        const int   c  = __float2int_rn(xv * 255.0f);
        const int   lo = max(c - RADIUS, 0);
        const int   hi = min(c + RADIUS, N_BINS - 1);
        for (int b = lo; b <= hi; ++b) {
            const float d = xv - (float)b * INV255;
            const float w = __builtin_amdgcn_exp2f(-GAUSS_K * d * d);  // v_exp_f32
            atomicAdd(&hs[b], w);                                      // ds_add_f32
        }
    }
    __syncthreads();
    atomicAdd(&ghist[batch * N_BINS + threadIdx.x], hs[threadIdx.x]);  // global_atomic_add_f32
}

// ---------------------------------------------------------------------------
// Kernel 2: normalize + cumsum + cdf-normalize. One wave (32 thr) per batch.
// Cumsum of 256 values = (16x16 H) x (16x16 upper-tri ones) via fp32 WMMA,
// then exclusive row-offset fixup. EXEC is all-ones (no divergence pre-WMMA).
// ---------------------------------------------------------------------------
__global__ __launch_bounds__(32) void cdf_kernel(const float* __restrict__ ghist,
                                                 float* __restrict__ cdfn) {
    const int batch = blockIdx.x;
    const int lane  = threadIdx.x;          // 0..31
    const float* h  = ghist + batch * N_BINS;

    // Total mass (wave butterfly reduction).
    float local = 0.0f;
#pragma unroll
    for (int j = 0; j < 8; ++j) local += h[lane + 32 * j];
#pragma unroll
    for (int o = 16; o > 0; o >>= 1) local += __shfl_xor(local, o, 32);
    const float invTot = 1.0f / (local + EPS);

    // A-fragment mapping (ISA 7.12.2, 32-bit A 16x4):
    //   lane L: M = L&15 ; VGPR0 holds K = 2*(L>>4), VGPR1 holds K+1.
    const int r  = lane & 15;               // row (A) / column (B,C)
    const int hi = lane >> 4;               // lane half

    v8f acc = {};
#pragma unroll
    for (int k0 = 0; k0 < 16; k0 += 4) {
        const int ka = k0 + (hi << 1);
        v2f a, b;
        a.x = h[r * 16 + ka] * invTot;      // H[r][ka]
        a.y = h[r * 16 + ka + 1] * invTot;  // H[r][ka+1]
        // B = upper-triangular ones chunk: B[k][n] = (k <= n), n = lane&15.
        b.x = (ka <= r) ? 1.0f : 0.0f;
        b.y = (ka + 1 <= r) ? 1.0f : 0.0f;
        acc = __builtin_amdgcn_wmma_f32_16x16x4_f32(
            /*neg_a=*/false, a, /*neg_b=*/false, b,
            /*c_mod=*/(short)0, acc, /*reuse_a=*/false, /*reuse_b=*/false);
    }

    // C layout: acc[j] = P[j + 8*hi][lane&15] ; P[row][n] = sum_{k<=n} H[row][k].
    __shared__ float P[N_BINS];
#pragma unroll
    for (int j = 0; j < 8; ++j) P[(j + (hi << 3)) * 16 + r] = acc[j];
    __syncthreads();

    // cdf[i] = P[i] + sum of full-row totals before row(i); then normalize.
    const float c0 = P[0];                  // cdf[0] == hn[0]
#pragma unroll
    for (int j = 0; j < 8; ++j) {
        const int i  = lane + 32 * j;
        const int rr = i >> 4;
        float off = 0.0f;
        for (int q = 0; q < rr; ++q) off += P[q * 16 + 15];
        const float cdf = P[i] + off;
        cdfn[batch * N_BINS + i] = (cdf - c0) / (1.0f - c0 + EPS);
    }
}

// ---------------------------------------------------------------------------
// Kernel 3: per-pixel equalization. grid = (blocksX, 2 batches), 256 thr.
// cdfn table staged into LDS via the gfx1250 async global->LDS path.
// ---------------------------------------------------------------------------
__global__ __launch_bounds__(256) void eq_kernel(const float* __restrict__ x,
                                                 const float* __restrict__ cdfn,
                                                 float* __restrict__ out,
                                                 int npix) {
    const int batch = blockIdx.y;
    __shared__ float cs[N_BINS];

    {
        // Async copy: one dword per thread. GVS mode: saddr = table base (SGPR
        // pair), vaddr = per-lane byte offset, vdst = raw LDS byte offset
        // (low 32 bits of the generic shared pointer == wave-relative LDS
        // address, ISA sec. 10.2 aperture mapping).
        const float* src     = cdfn + batch * N_BINS;   // uniform -> SGPRs
        unsigned     goff    = (unsigned)threadIdx.x * 4u;
        unsigned     lds_dst = (unsigned)(unsigned long long)&cs[threadIdx.x];
        asm volatile("global_load_async_to_lds_b32 %0, %1, %2"
                     :
                     : "v"(lds_dst), "v"(goff), "s"(src)
                     : "memory");
        asm volatile("s_wait_asynccnt 0" ::: "memory");   // my wave's copy done
    }
    __syncthreads();                                      // all waves' copies done

    const float* xb = x + (size_t)batch * (size_t)npix;
    float*       ob = out + (size_t)batch * (size_t)npix;
    const int idx    = blockIdx.x * blockDim.x + threadIdx.x;
    const int stride = gridDim.x * blockDim.x;

    for (int i = idx; i < npix; i += stride) {
        if (i + stride < npix) __builtin_prefetch(&xb[i + stride], 0, 0);  // global_prefetch_b8
        const float xv = xb[i];
        const int   c  = __float2int_rn(xv * 255.0f);
        const int   lo = max(c - RADIUS, 0);
        const int   hi = min(c + RADIUS, N_BINS - 1);
        float s = 0.0f, t = 0.0f;
        for (int b = lo; b <= hi; ++b) {
            const float d = xv - (float)b * INV255;
            const float w = __builtin_amdgcn_exp2f(-GAUSS_K * d * d);  // v_exp_f32
            s += w;
            t += w * cs[b];
        }
        ob[i] = t / (s + EPS);   // (sum w*cdfn) / (sum w + eps)
    }
}

// ---------------------------------------------------------------------------
extern "C" void kernel_launch(void* const* d_in, const int* in_sizes, int n_in,
                              void* d_out, int out_size, void* d_ws, size_t ws_size,
                              hipStream_t stream) {
    const float* x   = (const float*)d_in[0];
    float*       out = (float*)d_out;

    const int batches = 2;
    const int npix    = in_sizes[0] / batches;   // 512*512

    float* ghist = (float*)d_ws;                 // [2][256]
    float* cdfn  = ghist + batches * N_BINS;     // [2][256]

    hipMemsetAsync(ghist, 0, batches * N_BINS * sizeof(float), stream);

    dim3 grid(256, batches), block(256);
    hist_kernel<<<grid, block, 0, stream>>>(x, ghist, npix);
    cdf_kernel<<<dim3(batches), dim3(32), 0, stream>>>(ghist, cdfn);
    eq_kernel<<<grid, block, 0, stream>>>(x, cdfn, out, npix);
}